// QP_Layer_41850161332260
// MI455X (gfx1250) — compile-verified
//
#include <hip/hip_runtime.h>
#include <math.h>

// ---------------------------------------------------------------------------
// Batched PDIPM QP layer for MI455X (gfx1250, wave32).
// One wave32 workgroup per batch element. Exploits the fixed sparse G:
// H = Q + G' diag(w) G is block-[tridiag | diag | diag]; Schur-eliminate the
// two diagonal slack blocks -> 64x64 SPD system, factored with a WMMA-blocked
// Cholesky (v_wmma_f32_16x16x4_f32 trailing updates), reused for the
// predictor and corrector solves of each Mehrotra iteration.
// ---------------------------------------------------------------------------

typedef float v2f __attribute__((ext_vector_type(2)));
typedef float v8f __attribute__((ext_vector_type(8)));

#define NG     64      // n generators
#define NVAR   192     // 3n decision vars
#define MC     446     // 7n-2 constraints
#define MPAD   448
#define LDA    65      // 64x64 Schur matrix row stride (conflict-free cols)
#define EPSQ   1e-6f
#define CRAMP  0.5f
#define NITER  18

// ---------------- wave/WG reductions through LDS (single wave32 WG) --------
__device__ __forceinline__ float wg_sum(float v, volatile float* red, int tid) {
  red[tid] = v; __syncthreads();
  if (tid < 16) red[tid] += red[tid + 16]; __syncthreads();
  if (tid < 8)  red[tid] += red[tid + 8];  __syncthreads();
  if (tid < 4)  red[tid] += red[tid + 4];  __syncthreads();
  if (tid < 2)  red[tid] += red[tid + 2];  __syncthreads();
  if (tid < 1)  red[tid] += red[tid + 1];  __syncthreads();
  float r = red[0]; __syncthreads();
  return r;
}
__device__ __forceinline__ float wg_min(float v, volatile float* red, int tid) {
  red[tid] = v; __syncthreads();
  if (tid < 16) red[tid] = fminf(red[tid], red[tid + 16]); __syncthreads();
  if (tid < 8)  red[tid] = fminf(red[tid], red[tid + 8]);  __syncthreads();
  if (tid < 4)  red[tid] = fminf(red[tid], red[tid + 4]);  __syncthreads();
  if (tid < 2)  red[tid] = fminf(red[tid], red[tid + 2]);  __syncthreads();
  if (tid < 1)  red[tid] = fminf(red[tid], red[tid + 1]);  __syncthreads();
  float r = red[0]; __syncthreads();
  return r;
}

// ---------------- sparse G helpers -----------------------------------------
// Row m of G applied to (z,s,t):
__device__ __forceinline__ float g_row(const float* z1, const float* z2,
                                       const float* z3, int idx) {
  if (idx < 64)  return -z1[idx] - z2[idx];
  if (idx < 128) { int i = idx - 64;  return  z1[i] - z3[i]; }
  if (idx < 192) return -z1[idx - 128];
  if (idx < 256) return -z2[idx - 192];
  if (idx < 320) return -z3[idx - 256];
  if (idx < 383) { int j = idx - 320; return  z1[j + 1] - z1[j]; }
  { int j = idx - 383; return z1[j] - z1[j + 1]; }
}
// (G' q) components at generator index i:
__device__ __forceinline__ void gt_apply(const float* q, int i,
                                         float& gz, float& gs, float& gt) {
  float a = -q[i] + q[64 + i] - q[128 + i];
  if (i < 63) a += -q[320 + i] + q[383 + i];
  if (i > 0)  a +=  q[320 + i - 1] - q[383 + i - 1];
  gz = a;
  gs = -q[i] - q[192 + i];
  gt = -q[64 + i] - q[256 + i];
}

// ---------------- 64x64 WMMA-blocked Cholesky ------------------------------
__device__ void cholesky64(float* A, int tid) {
  const int half = tid >> 4;
  const int l15  = tid & 15;
  for (int kb = 0; kb < 4; ++kb) {
    const int b0 = kb * 16;
    // diagonal 16x16 block (scalar, column-sequential)
    for (int c = 0; c < 16; ++c) {
      if (tid == 0) {
        float d = A[(b0 + c) * LDA + b0 + c];
        for (int t = 0; t < c; ++t) { float l = A[(b0 + c) * LDA + b0 + t]; d -= l * l; }
        A[(b0 + c) * LDA + b0 + c] = sqrtf(fmaxf(d, 1e-30f));
      }
      __syncthreads();
      int r = c + 1 + tid;
      if (r < 16) {
        float v = A[(b0 + r) * LDA + b0 + c];
        for (int t = 0; t < c; ++t)
          v -= A[(b0 + r) * LDA + b0 + t] * A[(b0 + c) * LDA + b0 + t];
        A[(b0 + r) * LDA + b0 + c] = v / A[(b0 + c) * LDA + b0 + c];
      }
      __syncthreads();
    }
    // panel rows below: L[i,kb] = A[i,kb] * Lkk^{-T}  (row-parallel)
    const int nr = (3 - kb) * 16;
    for (int rr = tid; rr < nr; rr += 32) {
      const int r = b0 + 16 + rr;
      for (int c = 0; c < 16; ++c) {
        float v = A[r * LDA + b0 + c];
        for (int t = 0; t < c; ++t)
          v -= A[r * LDA + b0 + t] * A[(b0 + c) * LDA + b0 + t];
        A[r * LDA + b0 + c] = v / A[(b0 + c) * LDA + b0 + c];
      }
    }
    __syncthreads();
    // trailing update via WMMA: A[ib,jb] -= L[ib,kb] * L[jb,kb]^T
    for (int jb = kb + 1; jb < 4; ++jb) {
      for (int ib = jb; ib < 4; ++ib) {
        const int Ri = ib * 16, Rj = jb * 16, Ck = b0, Cj = jb * 16;
        v8f c;
        #pragma unroll
        for (int r = 0; r < 8; ++r)
          c[r] = A[(Ri + r + 8 * half) * LDA + Cj + l15];
        #pragma unroll
        for (int t = 0; t < 4; ++t) {
          const int cb = Ck + 4 * t + 2 * half;
          v2f a, b;
          a.x = -A[(Ri + l15) * LDA + cb];       // negate A-frag => C - L*L^T
          a.y = -A[(Ri + l15) * LDA + cb + 1];
          b.x =  A[(Rj + l15) * LDA + cb];       // B = L[jb,kb]^T in 4x16 layout
          b.y =  A[(Rj + l15) * LDA + cb + 1];
          c = __builtin_amdgcn_wmma_f32_16x16x4_f32(
                false, a, false, b, (short)0, c, false, false);
        }
        #pragma unroll
        for (int r = 0; r < 8; ++r)
          A[(Ri + r + 8 * half) * LDA + Cj + l15] = c[r];
        __syncthreads();
      }
    }
  }
}

// In-place solve L L^T x = b using factored A (lane-parallel column sweeps)
__device__ void chol_solve64(const float* A, float* b, int tid) {
  for (int j = 0; j < 64; ++j) {                 // forward: L y = b
    if (tid == 0) b[j] = b[j] / A[j * LDA + j];
    __syncthreads();
    float xj = b[j];
    int r = tid;       if (r  > j) b[r]  -= A[r  * LDA + j] * xj;
    int r2 = tid + 32; if (r2 > j) b[r2] -= A[r2 * LDA + j] * xj;
    __syncthreads();
  }
  for (int j = 63; j >= 0; --j) {                // backward: L^T x = y
    if (tid == 0) b[j] = b[j] / A[j * LDA + j];
    __syncthreads();
    float xj = b[j];
    int r = tid;       if (r  < j) b[r]  -= A[j * LDA + r ] * xj;
    int r2 = tid + 32; if (r2 < j) b[r2] -= A[j * LDA + r2] * xj;
    __syncthreads();
  }
}

// ---------------- main kernel ----------------------------------------------
__global__ __launch_bounds__(32)
void pdipm_qp_kernel(const float* __restrict__ yg,
                     const float* __restrict__ gu,
                     const float* __restrict__ go,
                     float* __restrict__ out) {
  __shared__ float Ash[64 * LDA];
  __shared__ float yL[64], guL[64], goL[64];
  __shared__ float zv[64], sv[64], tv[64];
  __shared__ float dzz[64], dzs[64], dzt[64];
  __shared__ float rdz[64], rds[64], rdt[64];
  __shared__ float rsv[64], rtv[64];
  __shared__ float Sd[64], Td[64];
  __shared__ float bv[64];
  __shared__ float sl[MPAD], lam[MPAD], rp[MPAD], rc[MPAD];
  __shared__ float dsv[MPAD], dlv[MPAD], wv[MPAD], tq[MPAD];
  __shared__ float red[32];

  const int tid = threadIdx.x;
  const int bat = blockIdx.x;

  // ---- load & init ----
  for (int ii = 0; ii < 2; ++ii) {
    int i = tid + 32 * ii;
    float yi = yg[bat * NG + i];
    yL[i] = yi; guL[i] = gu[i]; goL[i] = go[i];
    zv[i] = yi;                       // z0 = -Q^{-1} p
    sv[i] = -gu[i] / EPSQ;
    tv[i] = -go[i] / EPSQ;
  }
  __syncthreads();
  for (int jj = 0; jj < 14; ++jj) {
    int idx = tid + 32 * jj;
    if (idx < MC) {
      float h = (idx < 64) ? -yL[idx] : (idx < 128) ? yL[idx - 64]
                : (idx < 320) ? 0.f : CRAMP;
      sl[idx]  = fmaxf(h - g_row(zv, sv, tv, idx), 1.0f);
      lam[idx] = 1.0f;
    } else if (idx < MPAD) {
      sl[idx] = 1.f; lam[idx] = 0.f; rp[idx] = 0.f; rc[idx] = 0.f;
      dsv[idx] = 0.f; dlv[idx] = 0.f; wv[idx] = 0.f; tq[idx] = 0.f;
    }
  }
  __syncthreads();

  #pragma unroll 1
  for (int it = 0; it < NITER; ++it) {
    // ---- w = clip(lam/s), Schur diagonals ----
    for (int jj = 0; jj < 14; ++jj) {
      int idx = tid + 32 * jj;
      if (idx < MC) wv[idx] = fminf(fmaxf(lam[idx] / sl[idx], 1e-8f), 1e8f);
    }
    __syncthreads();
    for (int ii = 0; ii < 2; ++ii) {
      int i = tid + 32 * ii;
      Sd[i] = EPSQ + wv[i] + wv[192 + i];
      Td[i] = EPSQ + wv[64 + i] + wv[256 + i];
    }
    __syncthreads();

    // ---- build Schur matrix (tridiag + rank-correct diag) ----
    for (int k = tid; k < 64 * LDA; k += 32) Ash[k] = 0.f;
    __syncthreads();
    for (int ii = 0; ii < 2; ++ii) {
      int i = tid + 32 * ii;
      float w1 = wv[i], w2 = wv[64 + i], w3 = wv[128 + i];
      float d = 1.f + w1 + w2 + w3;
      if (i > 0)  d += wv[320 + i - 1] + wv[383 + i - 1];
      if (i < 63) d += wv[320 + i] + wv[383 + i];
      d -= w1 * w1 / Sd[i] + w2 * w2 / Td[i];
      Ash[i * LDA + i] = d;
      if (i < 63) {
        float e = -(wv[320 + i] + wv[383 + i]);
        Ash[i * LDA + i + 1] = e;
        Ash[(i + 1) * LDA + i] = e;
      }
    }
    __syncthreads();
    cholesky64(Ash, tid);             // WMMA-blocked factor, reused twice

    // ---- residuals r_d, r_p, mu ----
    for (int ii = 0; ii < 2; ++ii) {
      int i = tid + 32 * ii;
      float gz, gs, gt;
      gt_apply(lam, i, gz, gs, gt);
      rdz[i] = zv[i] - yL[i] + gz;
      rds[i] = EPSQ * sv[i] + guL[i] + gs;
      rdt[i] = EPSQ * tv[i] + goL[i] + gt;
    }
    float psum = 0.f;
    for (int jj = 0; jj < 14; ++jj) {
      int idx = tid + 32 * jj;
      if (idx < MC) {
        float h = (idx < 64) ? -yL[idx] : (idx < 128) ? yL[idx - 64]
                  : (idx < 320) ? 0.f : CRAMP;
        rp[idx] = g_row(zv, sv, tv, idx) + sl[idx] - h;
        psum += sl[idx] * lam[idx];
      }
    }
    __syncthreads();
    float mu = fmaxf(wg_sum(psum, red, tid) / (float)MC, 1e-12f);

    // ================= two solves sharing the factor =================
    for (int pass = 0; pass < 2; ++pass) {
      // r_c
      if (pass == 0) {
        for (int jj = 0; jj < 14; ++jj) {
          int idx = tid + 32 * jj;
          if (idx < MC) rc[idx] = sl[idx] * lam[idx];
        }
      }
      __syncthreads();

      // tq = (lam*r_p - r_c)/s ; rhs = -(r_d + G' tq)
      for (int jj = 0; jj < 14; ++jj) {
        int idx = tid + 32 * jj;
        if (idx < MC) tq[idx] = (lam[idx] * rp[idx] - rc[idx]) / sl[idx];
      }
      __syncthreads();
      for (int ii = 0; ii < 2; ++ii) {
        int i = tid + 32 * ii;
        float gz, gs, gt;
        gt_apply(tq, i, gz, gs, gt);
        float rz = -(rdz[i] + gz);
        float rs = -(rds[i] + gs);
        float rt = -(rdt[i] + gt);
        rsv[i] = rs; rtv[i] = rt;
        bv[i] = rz - wv[i] * rs / Sd[i] + wv[64 + i] * rt / Td[i];
      }
      __syncthreads();
      chol_solve64(Ash, bv, tid);
      for (int ii = 0; ii < 2; ++ii) {
        int i = tid + 32 * ii;
        dzz[i] = bv[i];
        dzs[i] = (rsv[i] - wv[i] * bv[i]) / Sd[i];
        dzt[i] = (rtv[i] + wv[64 + i] * bv[i]) / Td[i];
      }
      __syncthreads();
      float mn = 3.4e38f;
      for (int jj = 0; jj < 14; ++jj) {
        int idx = tid + 32 * jj;
        if (idx < MC) {
          float d = -rp[idx] - g_row(dzz, dzs, dzt, idx);
          float l = (-rc[idx] - lam[idx] * d) / sl[idx];
          dsv[idx] = d; dlv[idx] = l;
          if (d < 0.f) mn = fminf(mn, -sl[idx] / d);
          if (l < 0.f) mn = fminf(mn, -lam[idx] / l);
        }
      }
      __syncthreads();
      float a = fminf(1.f, 0.99f * wg_min(mn, red, tid));

      if (pass == 0) {
        // sigma from affine step; build corrector r_c
        float msum = 0.f;
        for (int jj = 0; jj < 14; ++jj) {
          int idx = tid + 32 * jj;
          if (idx < MC)
            msum += (sl[idx] + a * dsv[idx]) * (lam[idx] + a * dlv[idx]);
        }
        float mu_aff = wg_sum(msum, red, tid) / (float)MC;
        float ratio = mu_aff / mu;
        float sigma = ratio * ratio * ratio;
        for (int jj = 0; jj < 14; ++jj) {
          int idx = tid + 32 * jj;
          if (idx < MC)
            rc[idx] = sl[idx] * lam[idx] + dsv[idx] * dlv[idx] - sigma * mu;
        }
        __syncthreads();
      } else {
        // take the corrector step
        for (int ii = 0; ii < 2; ++ii) {
          int i = tid + 32 * ii;
          zv[i] += a * dzz[i];
          sv[i] += a * dzs[i];
          tv[i] += a * dzt[i];
        }
        for (int jj = 0; jj < 14; ++jj) {
          int idx = tid + 32 * jj;
          if (idx < MC) {
            sl[idx]  += a * dsv[idx];
            lam[idx] += a * dlv[idx];
          }
        }
        __syncthreads();
      }
    }
  }

  // ---- emit z* = (z, s, t) ----
  for (int ii = 0; ii < 2; ++ii) {
    int i = tid + 32 * ii;
    out[bat * NVAR + i]        = zv[i];
    out[bat * NVAR + 64 + i]   = sv[i];
    out[bat * NVAR + 128 + i]  = tv[i];
  }
}

extern "C" void kernel_launch(void* const* d_in, const int* in_sizes, int n_in,
                              void* d_out, int out_size, void* d_ws, size_t ws_size,
                              hipStream_t stream) {
  const float* y  = (const float*)d_in[0];
  const float* gu = (const float*)d_in[1];
  const float* go = (const float*)d_in[2];
  float* out = (float*)d_out;
  const int B = in_sizes[0] / NG;     // 256 batches
  pdipm_qp_kernel<<<B, 32, 0, stream>>>(y, gu, go, out);
}